// NGCFLayer_13408887898544
// MI455X (gfx1250) — compile-verified
//
#include <hip/hip_runtime.h>

typedef __attribute__((ext_vector_type(2))) float v2f;
typedef __attribute__((ext_vector_type(8))) float v8f;

#define DIM 128
#define LDSP 132   // padded LDS row stride (floats) to avoid bank conflicts
#define NEG_SLOPE 0.2f
#define EPS 1e-12f

// ---------------------------------------------------------------------------
// Kernel 1: zero the lap accumulator in workspace (float4 stores)
// ---------------------------------------------------------------------------
__global__ void zero_f32(float* __restrict__ p, int n4) {
  int i = blockIdx.x * blockDim.x + threadIdx.x;
  if (i < n4) {
    float4 z = make_float4(0.f, 0.f, 0.f, 0.f);
    reinterpret_cast<float4*>(p)[i] = z;
  }
}

// ---------------------------------------------------------------------------
// Kernel 2: COO SpMM via atomics. One wave (32 lanes) per edge; each lane
// handles 4 consecutive columns (float4 gather, 4x global_atomic_add_f32).
// ego table + lap accumulator are L2-resident (51.2 MB each, 192 MB L2).
// ---------------------------------------------------------------------------
__global__ void spmm_scatter(const float* __restrict__ ego,
                             const float* __restrict__ vals,
                             const int* __restrict__ src,
                             const int* __restrict__ dst,
                             float* __restrict__ lap, int E) {
  long gid = (long)blockIdx.x * blockDim.x + threadIdx.x;
  int e = (int)(gid >> 5);
  int lane = (int)(gid & 31);
  if (e >= E) return;
  int s = src[e];               // uniform within the wave -> scalar-cached
  int d = dst[e];
  float a = vals[e];
  const float4 v = reinterpret_cast<const float4*>(ego + (long)s * DIM)[lane];
  float* base = lap + (long)d * DIM + lane * 4;
  atomicAdd(base + 0, v.x * a);
  atomicAdd(base + 1, v.y * a);
  atomicAdd(base + 2, v.z * a);
  atomicAdd(base + 3, v.w * a);
}

// ---------------------------------------------------------------------------
// Kernel 3: fused  Y = leakyrelu((ego+lap)@W1 + (ego*lap)@W2 + b1 + b2),
//                  out = Y / max(||Y||_row, eps)
// Block = 128 threads (4 waves) handles 16 rows. Each wave owns two 16x16
// output tiles; both GEMMs accumulate into one v8f via V_WMMA_F32_16X16X4_F32.
// ---------------------------------------------------------------------------
__global__ __launch_bounds__(128) void ngcf_fused(
    const float* __restrict__ ego, const float* __restrict__ lap,
    const float* __restrict__ W1, const float* __restrict__ W2,
    const float* __restrict__ b1, const float* __restrict__ b2,
    float* __restrict__ out) {
  __shared__ float sX1[16 * LDSP];
  __shared__ float sX2[16 * LDSP];
  __shared__ float sY [16 * LDSP];
  __shared__ float sRn[16];

  const int tid  = threadIdx.x;
  const int row0 = blockIdx.x * 16;

  // Stage X1 = ego + lap, X2 = ego * lap for this 16-row tile.
  #pragma unroll 4
  for (int r = 0; r < 16; ++r) {
    float e = ego[(row0 + r) * DIM + tid];
    float l = lap[(row0 + r) * DIM + tid];
    sX1[r * LDSP + tid] = e + l;
    sX2[r * LDSP + tid] = e * l;
  }
  __syncthreads();

  const int wave = tid >> 5;
  const int lane = tid & 31;
  const int half = lane >> 4;    // 0: lanes 0-15, 1: lanes 16-31
  const int lr   = lane & 15;

  for (int t = 0; t < 2; ++t) {
    const int nb  = (wave * 2 + t) * 16;   // output column tile base
    const int col = nb + lr;
    v8f c = {};
    #pragma unroll 8
    for (int k = 0; k < DIM; k += 4) {
      const int ka = k + 2 * half;         // K indices {ka, ka+1} for this lane half
      // A fragments: 16x4 f32, lane half selects K pair (ISA 7.12.2)
      v2f a1 = *reinterpret_cast<const v2f*>(&sX1[lr * LDSP + ka]);
      v2f a2 = *reinterpret_cast<const v2f*>(&sX2[lr * LDSP + ka]);
      // B fragments: 4x16 f32, row-striped across lanes, K pair per half
      v2f bb1, bb2;
      bb1.x = W1[ka * DIM + col];
      bb1.y = W1[(ka + 1) * DIM + col];
      bb2.x = W2[ka * DIM + col];
      bb2.y = W2[(ka + 1) * DIM + col];
      c = __builtin_amdgcn_wmma_f32_16x16x4_f32(false, a1, false, bb1,
                                                (short)0, c, false, false);
      c = __builtin_amdgcn_wmma_f32_16x16x4_f32(false, a2, false, bb2,
                                                (short)0, c, false, false);
    }
    // Epilogue per tile: bias + leaky relu, park in LDS for row norms.
    const float bias = b1[col] + b2[col];
    #pragma unroll
    for (int j = 0; j < 8; ++j) {
      float y = c[j] + bias;
      y = (y > 0.f) ? y : NEG_SLOPE * y;
      sY[(j + 8 * half) * LDSP + col] = y;   // row = j + 8*half, per C/D layout
    }
  }
  __syncthreads();

  // Row L2 norms across all 128 columns.
  if (tid < 16) {
    float ss = 0.f;
    #pragma unroll 8
    for (int cIdx = 0; cIdx < DIM; ++cIdx) {
      float y = sY[tid * LDSP + cIdx];
      ss += y * y;
    }
    float nrm = sqrtf(ss);
    sRn[tid] = 1.0f / fmaxf(nrm, EPS);
  }
  __syncthreads();

  #pragma unroll 4
  for (int r = 0; r < 16; ++r) {
    out[(row0 + r) * DIM + tid] = sY[r * LDSP + tid] * sRn[r];
  }
}

// ---------------------------------------------------------------------------
extern "C" void kernel_launch(void* const* d_in, const int* in_sizes, int n_in,
                              void* d_out, int out_size, void* d_ws, size_t ws_size,
                              hipStream_t stream) {
  const float* ego = (const float*)d_in[0];
  const float* W1  = (const float*)d_in[1];
  const float* W2  = (const float*)d_in[2];
  const float* b1  = (const float*)d_in[3];
  const float* b2  = (const float*)d_in[4];
  const float* av  = (const float*)d_in[5];
  const int*   src = (const int*)d_in[6];
  const int*   dst = (const int*)d_in[7];
  float* out = (float*)d_out;
  float* lap = (float*)d_ws;          // N*DIM floats of scratch

  const int N = in_sizes[0] / DIM;    // 100000
  const int E = in_sizes[5];          // 1600000
  const int nElems = N * DIM;

  // 1) zero lap accumulator
  {
    int n4 = nElems / 4;
    zero_f32<<<(n4 + 255) / 256, 256, 0, stream>>>(lap, n4);
  }
  // 2) SpMM scatter: one wave per edge
  {
    long threads = (long)E * 32;
    long blocks = (threads + 255) / 256;
    spmm_scatter<<<(int)blocks, 256, 0, stream>>>(ego, av, src, dst, lap, E);
  }
  // 3) fused dual-GEMM + bias + leakyrelu + L2-normalize (N % 16 == 0)
  ngcf_fused<<<N / 16, 128, 0, stream>>>(ego, lap, W1, W2, b1, b2, out);
}